// GAT_71846212927557
// MI455X (gfx1250) — compile-verified
//
#include <hip/hip_runtime.h>
#include <math.h>

typedef float v2f __attribute__((ext_vector_type(2)));
typedef float v8f __attribute__((ext_vector_type(8)));

// ---------------------------------------------------------------- utilities
__global__ void k_fill(float* __restrict__ p, long long n, float v) {
    long long i = (long long)blockIdx.x * blockDim.x + threadIdx.x;
    if (i < n) p[i] = v;
}

// sign-split float atomic max (correct with init = -inf)
__device__ inline void atomicMaxF(float* addr, float v) {
    if (v >= 0.0f) atomicMax((int*)addr, __float_as_int(v));
    else           atomicMin((unsigned int*)addr, __float_as_uint(v));
}

// ------------------------------------------------- WMMA GEMM: OUT = X @ W
// X: [N,K] row-major, W: [K,M] row-major, OUT: [N,M].  K % 4 == 0, M % 16 == 0.
// 4 waves per block; each wave computes one 16x16 tile with f32 WMMA (exact).
// The in-bounds / tail decision is wave-uniform, so EXEC is all-ones around
// every WMMA in both paths.
__global__ void k_gemm_wmma(const float* __restrict__ X, const float* __restrict__ W,
                            float* __restrict__ OUT, int N, int K, int M) {
    const int lane = threadIdx.x & 31;
    const int wave = threadIdx.x >> 5;
    const int tile = blockIdx.x * 4 + wave;
    const int r0   = tile * 16;
    const int n0   = blockIdx.y * 16;
    const int hi   = lane >> 4;      // 0: K pair (k0,k0+1)   1: (k0+2,k0+3)
    const int l15  = lane & 15;
    const float* __restrict__ Wb = W + n0 + l15;

    v8f c = {};
#if __has_builtin(__builtin_amdgcn_wmma_f32_16x16x4_f32)
    if (r0 + 16 <= N) {
        // ---- fast path: full tile, no masking ----
        const float* __restrict__ Xrow = X + (long long)(r0 + l15) * K + 2 * hi;
#pragma unroll 8
        for (int k0 = 0; k0 < K; k0 += 4) {
            v2f a;
            a.x = Xrow[k0];                  // A 16x4 layout: lane<16 -> K=k0,k0+1
            a.y = Xrow[k0 + 1];              //               lane>=16 -> K=k0+2,k0+3
            const int kb = k0 + 2 * hi;
            v2f b;
            b.x = Wb[(long long)kb * M];     // B 4x16: rows striped across lanes/VGPRs
            b.y = Wb[(long long)(kb + 1) * M];
            c = __builtin_amdgcn_wmma_f32_16x16x4_f32(
                    false, a, false, b, (short)0, c, false, false);
        }
#pragma unroll
        for (int j = 0; j < 8; ++j) {        // C/D: VGPR j -> M=j (+8 for upper lanes)
            OUT[(long long)(r0 + j + 8 * hi) * M + n0 + l15] = c[j];
        }
    } else if (r0 < N) {
        // ---- tail path: mask out-of-range rows ----
        const int  arow   = r0 + l15;
        const int  arow_c = arow < N ? arow : (N - 1);
        const float amask = arow < N ? 1.0f : 0.0f;
        const float* __restrict__ Xrow = X + (long long)arow_c * K + 2 * hi;
        for (int k0 = 0; k0 < K; k0 += 4) {
            v2f a;
            a.x = Xrow[k0]     * amask;
            a.y = Xrow[k0 + 1] * amask;
            const int kb = k0 + 2 * hi;
            v2f b;
            b.x = Wb[(long long)kb * M];
            b.y = Wb[(long long)(kb + 1) * M];
            c = __builtin_amdgcn_wmma_f32_16x16x4_f32(
                    false, a, false, b, (short)0, c, false, false);
        }
#pragma unroll
        for (int j = 0; j < 8; ++j) {
            int row = r0 + j + 8 * hi;
            if (row < N) OUT[(long long)row * M + n0 + l15] = c[j];
        }
    }
#else
    // scalar fallback (should not be taken on gfx1250)
    if (r0 < N) {
        for (int j = 0; j < 8; ++j) {
            int row = r0 + j + 8 * hi;
            if (row >= N) continue;
            float acc = 0.f;
            const float* xr = X + (long long)row * K;
            for (int k = 0; k < K; ++k) acc += xr[k] * Wb[(long long)k * M];
            OUT[(long long)row * M + n0 + l15] = acc;
        }
    }
#endif
}

// ------------------------------------------------- attention pieces
// asrc[n,h] = sum_c h[n,h,c]*a_src[h,c] ; adst likewise
__global__ void k_attn_coeff(const float* __restrict__ h, const float* __restrict__ a_s,
                             const float* __restrict__ a_d, float* __restrict__ asrc,
                             float* __restrict__ adst, int N, int H, int C) {
    int idx = blockIdx.x * blockDim.x + threadIdx.x;
    if (idx >= N * H) return;
    int n = idx / H, hh = idx - n * H;
    const float* hp = h + (long long)n * H * C + (long long)hh * C;
    const float* ws = a_s + hh * C;
    const float* wd = a_d + hh * C;
    float s1 = 0.f, s2 = 0.f;
    for (int c = 0; c < C; ++c) { float v = hp[c]; s1 += v * ws[c]; s2 += v * wd[c]; }
    asrc[idx] = s1; adst[idx] = s2;
}

// logits = leaky_relu(asrc[src]+adst[dst]); segment max over dst via atomics
__global__ void k_edge_logit(const int* __restrict__ src, const int* __restrict__ dst,
                             const float* __restrict__ asrc, const float* __restrict__ adst,
                             float* __restrict__ logits, float* __restrict__ m,
                             long long E, int H) {
    long long i = (long long)blockIdx.x * blockDim.x + threadIdx.x;
    if (i >= E * H) return;
    long long e = i / H; int hh = (int)(i - e * H);
    int s = src[e], d = dst[e];
    float l = asrc[(long long)s * H + hh] + adst[(long long)d * H + hh];
    l = l > 0.f ? l : 0.2f * l;
    logits[i] = l;
    atomicMaxF(&m[(long long)d * H + hh], l);
}

// ev = exp(logit - m[dst]); segment sum via atomics; ev stored in place
__global__ void k_edge_expsum(const int* __restrict__ dst, float* __restrict__ logits,
                              const float* __restrict__ m, float* __restrict__ ssum,
                              long long E, int H) {
    long long i = (long long)blockIdx.x * blockDim.x + threadIdx.x;
    if (i >= E * H) return;
    long long e = i / H; int hh = (int)(i - e * H);
    int d = dst[e];
    float ev = expf(logits[i] - m[(long long)d * H + hh]);
    logits[i] = ev;
    atomicAdd(&ssum[(long long)d * H + hh], ev);
}

// alpha = ev / (s[dst] + 1e-16), in place (becomes the a1/a2/a3 output)
__global__ void k_edge_alpha(const int* __restrict__ dst, float* __restrict__ logits,
                             const float* __restrict__ ssum, long long E, int H) {
    long long i = (long long)blockIdx.x * blockDim.x + threadIdx.x;
    if (i >= E * H) return;
    long long e = i / H; int hh = (int)(i - e * H);
    int d = dst[e];
    logits[i] = logits[i] / (ssum[(long long)d * H + hh] + 1e-16f);
}

// agg[dst, h, c] += h[src, h, c] * alpha[e, h]
// One edge per block (blockDim == H*C): src/dst become block-uniform scalar
// loads and the gather/atomic-add are fully coalesced.
__global__ void k_edge_scatter_block(const int* __restrict__ src, const int* __restrict__ dst,
                                     const float* __restrict__ h, const float* __restrict__ alpha,
                                     float* __restrict__ agg, int H, int C) {
    const long long e  = blockIdx.x;
    const int ch = threadIdx.x;          // 0 .. H*C-1
    const int hh = ch / C;
    const int s = src[e];                // uniform -> scalar load
    const int d = dst[e];                // uniform -> scalar load
    const long long HC = (long long)H * C;
    float a = alpha[e * H + hh];
    atomicAdd(&agg[(long long)d * HC + ch], h[(long long)s * HC + ch] * a);
}

// flat variant for small HC (layer 3, HC = 16)
__global__ void k_edge_scatter(const int* __restrict__ src, const int* __restrict__ dst,
                               const float* __restrict__ h, const float* __restrict__ alpha,
                               float* __restrict__ agg, long long E, int H, int C) {
    long long i  = (long long)blockIdx.x * blockDim.x + threadIdx.x;
    long long HC = (long long)H * C;
    if (i >= E * HC) return;
    long long e = i / HC; int rem = (int)(i - e * HC); int hh = rem / C;
    float a = alpha[e * H + hh];
    float v = h[(long long)src[e] * HC + rem] * a;
    atomicAdd(&agg[(long long)dst[e] * HC + rem], v);
}

__global__ void k_bias_act(const float* __restrict__ agg, const float* __restrict__ bias,
                           float* __restrict__ out, long long total, int F, int relu) {
    long long i = (long long)blockIdx.x * blockDim.x + threadIdx.x;
    if (i >= total) return;
    int c = (int)(i % F);
    float v = agg[i] + bias[c];
    out[i] = relu ? fmaxf(v, 0.f) : v;
}

// ------------------------------------------------- pooling + classifier
__global__ void k_pool_accum(const float* __restrict__ emb, const int* __restrict__ batch,
                             float* __restrict__ gsum, float* __restrict__ counts,
                             int N, int F) {
    long long i = (long long)blockIdx.x * blockDim.x + threadIdx.x;
    if (i >= (long long)N * F) return;
    int n = (int)(i / F), c = (int)(i - (long long)n * F);
    int g = batch[n];
    atomicAdd(&gsum[g * F + c], emb[i]);
    if (c == 0) atomicAdd(&counts[g], 1.0f);
}

__global__ void k_pool_div(float* gsum, const float* counts, int NG, int F) {
    int i = threadIdx.x;
    if (i >= NG * F) return;
    gsum[i] /= fmaxf(counts[i / F], 1.0f);
}

__global__ void k_classifier(const float* __restrict__ gsum, const float* __restrict__ Wl,
                             const float* __restrict__ bl, float* __restrict__ out,
                             int NG, int F, int O) {
    int i = threadIdx.x;
    if (i >= NG * O) return;
    int g = i / O, j = i - g * O;
    float acc = bl[j];
    for (int c = 0; c < F; ++c) acc += gsum[g * F + c] * Wl[c * O + j];
    out[i] = acc;
}

// ---------------------------------------------------------------- driver
extern "C" void kernel_launch(void* const* d_in, const int* in_sizes, int n_in,
                              void* d_out, int out_size, void* d_ws, size_t ws_size,
                              hipStream_t stream) {
    const float* x    = (const float*)d_in[0];
    const int*   ei   = (const int*)d_in[1];
    const int*   batch= (const int*)d_in[2];
    const float* W1 = (const float*)d_in[3];
    const float* as1= (const float*)d_in[4];
    const float* ad1= (const float*)d_in[5];
    const float* b1 = (const float*)d_in[6];
    const float* W2 = (const float*)d_in[7];
    const float* as2= (const float*)d_in[8];
    const float* ad2= (const float*)d_in[9];
    const float* b2 = (const float*)d_in[10];
    const float* W3 = (const float*)d_in[11];
    const float* as3= (const float*)d_in[12];
    const float* ad3= (const float*)d_in[13];
    const float* b3 = (const float*)d_in[14];
    const float* Wl = (const float*)d_in[15];
    const float* bl = (const float*)d_in[16];

    const int       FIN = 128;
    const int       N   = in_sizes[0] / FIN;
    const long long E   = in_sizes[1] / 2;
    const int NG = 16, EMB = 16, OUTC = 2;
    const int* src = ei;
    const int* dst = ei + E;

    // workspace layout (floats)
    float* wsf     = (float*)d_ws;
    float* h_buf   = wsf;                              // [N,128]
    float* agg_buf = h_buf   + (long long)N * 128;     // [N,128]
    float* post    = agg_buf + (long long)N * 128;     // [N,128]
    float* asrc    = post    + (long long)N * 128;     // [N,2]
    float* adst    = asrc    + (long long)N * 2;       // [N,2]
    float* mbuf    = adst    + (long long)N * 2;       // [N,2]
    float* sbuf    = mbuf    + (long long)N * 2;       // [N,2]
    float* counts  = sbuf    + (long long)N * 2;       // [16]
    (void)ws_size;

    // output layout: out | g_emb | emb | a1 | a2 | a3
    float* outp = (float*)d_out;
    float* gemb = outp + NG * OUTC;
    float* embp = gemb + NG * EMB;
    float* a1p  = embp + (long long)N * EMB;
    float* a2p  = a1p  + E * 2;
    float* a3p  = a2p  + E;
    (void)out_size; (void)n_in;

    const int TPB = 256;
    auto nblk = [](long long n, int tpb) { return (unsigned)((n + tpb - 1) / tpb); };

    auto layer = [&](const float* Xin, const float* Wm, const float* a_s, const float* a_d,
                     const float* bias, int H, int C, float* alpha_out, float* post_out,
                     int relu) {
        const int HC = H * C;
        const int rowTiles = (N + 15) / 16;
        dim3 grid((rowTiles + 3) / 4, HC / 16);
        k_gemm_wmma<<<grid, 128, 0, stream>>>(Xin, Wm, h_buf, N, FIN, HC);
        k_attn_coeff<<<nblk((long long)N * H, TPB), TPB, 0, stream>>>(
            h_buf, a_s, a_d, asrc, adst, N, H, C);
        k_fill<<<nblk((long long)N * H, TPB), TPB, 0, stream>>>(mbuf, (long long)N * H, -INFINITY);
        k_fill<<<nblk((long long)N * H, TPB), TPB, 0, stream>>>(sbuf, (long long)N * H, 0.f);
        k_fill<<<nblk((long long)N * HC, TPB), TPB, 0, stream>>>(agg_buf, (long long)N * HC, 0.f);
        k_edge_logit <<<nblk(E * H, TPB), TPB, 0, stream>>>(src, dst, asrc, adst, alpha_out, mbuf, E, H);
        k_edge_expsum<<<nblk(E * H, TPB), TPB, 0, stream>>>(dst, alpha_out, mbuf, sbuf, E, H);
        k_edge_alpha <<<nblk(E * H, TPB), TPB, 0, stream>>>(dst, alpha_out, sbuf, E, H);
        if (HC >= 64) {
            k_edge_scatter_block<<<(unsigned)E, HC, 0, stream>>>(
                src, dst, h_buf, alpha_out, agg_buf, H, C);
        } else {
            k_edge_scatter<<<nblk(E * HC, TPB), TPB, 0, stream>>>(
                src, dst, h_buf, alpha_out, agg_buf, E, H, C);
        }
        k_bias_act<<<nblk((long long)N * HC, TPB), TPB, 0, stream>>>(
            agg_buf, bias, post_out, (long long)N * HC, HC, relu);
    };

    // conv1: 128 -> 64 x 2 heads, concat, relu
    layer(x,    W1, as1, ad1, b1, 2, 64,  a1p, post, 1);
    // conv2: 128 -> 128, 1 head (mean == identity), relu
    layer(post, W2, as2, ad2, b2, 1, 128, a2p, post, 1);
    // conv3: 128 -> 16, 1 head, no relu -> emb output
    layer(post, W3, as3, ad3, b3, 1, 16,  a3p, embp, 0);

    // global mean pool + linear classifier
    k_fill<<<1, 256, 0, stream>>>(gemb, (long long)NG * EMB, 0.f);
    k_fill<<<1, 32,  0, stream>>>(counts, (long long)NG, 0.f);
    k_pool_accum<<<nblk((long long)N * EMB, TPB), TPB, 0, stream>>>(embp, batch, gemb, counts, N, EMB);
    k_pool_div<<<1, 256, 0, stream>>>(gemb, counts, NG, EMB);
    k_classifier<<<1, 32, 0, stream>>>(gemb, Wl, bl, outp, NG, EMB, OUTC);
}